// InnerConvexViolationProjection_56495999811643
// MI455X (gfx1250) — compile-verified
//
#include <hip/hip_runtime.h>

#define MDIM 64
#define NDIM 128
#define SROW 132          // padded LDS row stride (floats): 16B-aligned rows, TDM pads 4 DW per 128 DW
#define TROW 68           // transposed-copy row stride (floats), 16B-aligned
#define NTHREADS 128

typedef __attribute__((ext_vector_type(4))) unsigned int v4u;
typedef __attribute__((ext_vector_type(8))) int v8i;
typedef __attribute__((ext_vector_type(4))) int v4i;

__global__ __launch_bounds__(NTHREADS)
void icvp_kernel(const float* __restrict__ X,
                 const float* __restrict__ A,
                 const float* __restrict__ Bc,
                 const float* __restrict__ Mk,
                 float* __restrict__ Out)
{
    __shared__ __align__(16) float sA [MDIM * SROW];  // 33792 B, row-major A_w
    __shared__ __align__(16) float sAt[NDIM * TROW];  // 34816 B, column-major copy (A_w^T row-major)
    __shared__ __align__(16) float sx [NDIM];
    __shared__ __align__(16) float sd [NDIM];
    __shared__ __align__(16) float su [MDIM];
    __shared__ __align__(16) float sad[MDIM];
    __shared__ __align__(16) float sbt[MDIM];
    __shared__ __align__(16) float sbw[MDIM];
    __shared__ __align__(16) float sinv[MDIM];
    __shared__ __align__(16) float sred[NTHREADS];

    const int t  = threadIdx.x;
    const int bs = blockIdx.x;
    const float* Ap = A + (long long)bs * (MDIM * NDIM);

    // ---------------- Stage A tile into LDS (padded rows) via Tensor Data Mover ----------------
#if defined(__has_builtin) && __has_builtin(__builtin_amdgcn_tensor_load_to_lds)
    if (t < 32) {   // one wave issues the TDM op (EXEC ignored by tensor ops)
        unsigned      lds = (unsigned)(unsigned long long)(void*)sA;   // LDS byte offset
        unsigned long long ga = (unsigned long long)(void*)Ap;
        v4u g0; v8i g1; v4i g2; v4i g3; v8i g4;
        // D# group 0: count=1 | lds_addr | global_addr[56:0] | type=2
        g0[0] = 1u;
        g0[1] = lds;
        g0[2] = (unsigned)(ga & 0xffffffffull);
        g0[3] = (unsigned)((ga >> 32) & 0x01ffffffull) | (2u << 30);
        // D# group 1: data_size=4B | pad_enable | pad_interval=128 DW (code 6) | pad_amount=4 DW (code 3)
        g1[0] = (int)((2u << 16) | (1u << 20) | (6u << 22) | (3u << 25));
        g1[1] = (int)((unsigned)NDIM << 16);   // tensor_dim0 = 128 (bits 79:48, low part)
        g1[2] = (int)((unsigned)MDIM << 16);   // tensor_dim0 hi=0 | tensor_dim1 = 64 (low part)
        g1[3] = (int)((unsigned)NDIM << 16);   // tensor_dim1 hi=0 | tile_dim0 = 128
        g1[4] = MDIM;                          // tile_dim1 = 64, tile_dim2 = 0
        g1[5] = NDIM;                          // tensor_dim0_stride = 128 (low part)
        g1[6] = 0;
        g1[7] = 0;
        g2[0] = g2[1] = g2[2] = g2[3] = 0;     // 2D tensor: groups 2/3 unused
        g3[0] = g3[1] = g3[2] = g3[3] = 0;
        g4[0] = g4[1] = g4[2] = g4[3] = 0;
        g4[4] = g4[5] = g4[6] = g4[7] = 0;
        __builtin_amdgcn_tensor_load_to_lds(g0, g1, g2, g3, g4, 0);
        __builtin_amdgcn_s_wait_tensorcnt(0);
    }
#else
    for (int idx = t; idx < MDIM * NDIM; idx += NTHREADS)
        sA[(idx >> 7) * SROW + (idx & (NDIM - 1))] = Ap[idx];
#endif
    __syncthreads();

    // ---------------- helpers (all hot LDS traffic is b128) ----------------
    // outM = [relu]( A_w * vecN [- bsub] )   (row phase: 2 threads per row, float4 loads)
    auto matm = [&](const float* vecN, float* outM, const float* bsub, bool relu) {
        const int row  = t & (MDIM - 1);
        const int half = t >> 6;
        const float4* ap = reinterpret_cast<const float4*>(&sA[row * SROW + half * 64]);
        const float4* vp = reinterpret_cast<const float4*>(&vecN[half * 64]);
        float s0 = 0.f, s1 = 0.f, s2 = 0.f, s3 = 0.f;
        #pragma unroll
        for (int k = 0; k < 16; ++k) {
            float4 a = ap[k], v = vp[k];
            s0 = __builtin_fmaf(a.x, v.x, s0);
            s1 = __builtin_fmaf(a.y, v.y, s1);
            s2 = __builtin_fmaf(a.z, v.z, s2);
            s3 = __builtin_fmaf(a.w, v.w, s3);
        }
        sred[t] = (s0 + s1) + (s2 + s3);
        __syncthreads();
        if (t < MDIM) {
            float r = sred[t] + sred[t + MDIM];
            if (bsub) r -= bsub[t];
            if (relu) r = fmaxf(r, 0.0f);
            outM[t] = r;
        }
        __syncthreads();
    };
    // returns (A_w^T * vecM)[t]  (thread t owns row t of sAt, float4 loads)
    auto matT = [&](const float* vecM) -> float {
        const float4* ap = reinterpret_cast<const float4*>(&sAt[t * TROW]);
        const float4* vp = reinterpret_cast<const float4*>(vecM);
        float s0 = 0.f, s1 = 0.f, s2 = 0.f, s3 = 0.f;
        #pragma unroll
        for (int i = 0; i < 16; ++i) {
            float4 a = ap[i], v = vp[i];
            s0 = __builtin_fmaf(a.x, v.x, s0);
            s1 = __builtin_fmaf(a.y, v.y, s1);
            s2 = __builtin_fmaf(a.z, v.z, s2);
            s3 = __builtin_fmaf(a.w, v.w, s3);
        }
        return (s0 + s1) + (s2 + s3);
    };
    auto redSum = [&](float v) -> float {
        sred[t] = v; __syncthreads();
        for (int off = NTHREADS / 2; off > 0; off >>= 1) {
            if (t < off) sred[t] += sred[t + off];
            __syncthreads();
        }
        float r = sred[0]; __syncthreads();
        return r;
    };

    // ---------------- row-normalize constraints ----------------
    if (t < MDIM) {
        const float4* ap = reinterpret_cast<const float4*>(&sA[t * SROW]);
        float s0 = 0.f, s1 = 0.f, s2 = 0.f, s3 = 0.f;
        #pragma unroll
        for (int k = 0; k < 32; ++k) {
            float4 a = ap[k];
            s0 = __builtin_fmaf(a.x, a.x, s0);
            s1 = __builtin_fmaf(a.y, a.y, s1);
            s2 = __builtin_fmaf(a.z, a.z, s2);
            s3 = __builtin_fmaf(a.w, a.w, s3);
        }
        float nrm = fmaxf(sqrtf((s0 + s1) + (s2 + s3)), 1e-12f);
        float inv = 1.0f / nrm;
        sinv[t] = inv;
        float bw = Bc[(long long)bs * MDIM + t] * inv;
        sbw[t] = bw;
        sbt[t] = bw - 0.001f;               // MU_INSIDE
    }
    __syncthreads();
    {   // scale rows of A in place (128 threads, half-row each)
        const int row  = t & (MDIM - 1);
        const int half = t >> 6;
        float inv = sinv[row];
        float* ap = &sA[row * SROW + half * 64];
        #pragma unroll
        for (int k = 0; k < 64; ++k) ap[k] *= inv;
    }
    __syncthreads();
    {   // build transposed copy: sAt[c][i] = A_w[i][c]; thread t owns column c = t
        #pragma unroll
        for (int i = 0; i < MDIM; ++i) sAt[t * TROW + i] = sA[i * SROW + t];
    }
    __syncthreads();

    // ---------------- power iteration for eta ----------------
    sx[t] = 0.08838834764831845f;           // ones(128) normalized
    __syncthreads();
    #pragma unroll 1
    for (int pit = 0; pit < 5; ++pit) {
        matm(sx, su, nullptr, false);       // su = A_w v
        float atav = matT(su);              // (A^T A v)[t]
        float nn = sqrtf(redSum(atav * atav)) + 1e-12f;
        sx[t] = atav / nn;
        __syncthreads();
    }
    matm(sx, su, nullptr, false);           // Av with final v
    float avt = (t < MDIM) ? su[t] : 0.0f;
    const float eta = 0.99f / (redSum(avt * avt) + 1e-12f);   // ETA_MARGIN / (num + RHO)

    // ---------------- projected violation gradient descent ----------------
    const float mask = Mk[(long long)bs * NDIM + t];
    const float x_in = fmaxf(X[(long long)bs * NDIM + t], 0.0f);
    sx[t] = x_in;
    __syncthreads();

    #pragma unroll 1
    for (int it = 0; it < 30; ++it) {
        matm(sx, su, sbt, true);            // su = relu(A_w x - b_tight)
        float g  = matT(su);                // (A_w^T su)[t]
        float xn = fmaxf(sx[t] - eta * g, 0.0f) * mask;
        sx[t] = xn;
        __syncthreads();
    }

    // ---------------- alpha-map stage ----------------
    matm(sx, su, nullptr, false);           // su = A_w x_
    float viol = (t < MDIM) ? fmaxf(su[t] - sbw[t], 0.0f) : 0.0f;
    sred[t] = viol; __syncthreads();
    for (int off = NTHREADS / 2; off > 0; off >>= 1) {
        if (t < off) sred[t] = fmaxf(sred[t], sred[t + off]);
        __syncthreads();
    }
    const bool do_alpha = (sred[0] <= 1e-7f);   // ALPHA_FEAS_TOL
    __syncthreads();

    const float d_t = x_in * mask - sx[t];  // d = x_target - x_
    sd[t] = d_t;
    __syncthreads();
    matm(sd, sad, nullptr, false);          // sad = A_w d

    float ai = __builtin_inff();
    if (t < MDIM) {
        float Ad = sad[t];
        float slack = sbw[t] - su[t];
        ai = (Ad > 0.0f) ? (slack / (Ad + 1e-12f)) : __builtin_inff();
    }
    sred[t] = ai; __syncthreads();
    for (int off = NTHREADS / 2; off > 0; off >>= 1) {
        if (t < off) sred[t] = fminf(sred[t], sred[t + off]);
        __syncthreads();
    }
    float alpha = sred[0];
    if (!(alpha >= -3.0e38f && alpha <= 3.0e38f)) alpha = 1.0f;  // non-finite -> 1
    alpha = fminf(fmaxf(alpha - 1e-6f, 0.0f), 1.0f);             // clip(alpha - EPS, 0, 1)

    float xcur = sx[t];
    float xa   = fmaxf(xcur + alpha * d_t, 0.0f);
    float xf   = do_alpha ? xa : xcur;
    xf = fmaxf(xf, 0.0f) * mask;
    Out[(long long)bs * NDIM + t] = fmaxf(xf, 0.0f);
}

extern "C" void kernel_launch(void* const* d_in, const int* in_sizes, int n_in,
                              void* d_out, int out_size, void* d_ws, size_t ws_size,
                              hipStream_t stream) {
    (void)n_in; (void)out_size; (void)d_ws; (void)ws_size;
    const float* x  = (const float*)d_in[0];
    const float* A  = (const float*)d_in[1];
    const float* b  = (const float*)d_in[2];
    const float* mk = (const float*)d_in[3];
    float* out = (float*)d_out;
    const int nprob = in_sizes[2] / MDIM;   // B*S problems (b has m=64 per problem)
    icvp_kernel<<<nprob, NTHREADS, 0, stream>>>(x, A, b, mk, out);
}